// Teacher_S_84945863180994
// MI455X (gfx1250) — compile-verified
//
#include <hip/hip_runtime.h>
#include <hip/hip_bf16.h>
#include <stdint.h>

// Problem constants from the reference
static constexpr int NN   = 2048;   // graph nodes
static constexpr int IND  = 1024;   // input dim
static constexpr int HID  = 256;    // hidden dim after Wg1
static constexpr int H2   = 64;     // hidden dim after Wl2
static constexpr int NCLS = 4;

typedef __attribute__((ext_vector_type(2))) float v2f;
typedef __attribute__((ext_vector_type(4))) float v4f;
typedef __attribute__((ext_vector_type(8))) float v8f;

// ---------------------------------------------------------------------------
// CDNA5 async global->LDS copy (ASYNCcnt-tracked DMA, no VGPR staging).
//   the LDS destination is a byte offset (from the LDS base) held in the vdst VGPR; flat LDS pointers carry the wave-relative
//   offset in addr[31:0], so truncating a generic LDS pointer is the offset.
// ---------------------------------------------------------------------------
__device__ __forceinline__ uint32_t lds_off(const void* p) {
    return (uint32_t)(uintptr_t)p;
}
__device__ __forceinline__ void async_ld_b128(uint32_t lds_byte_off, const float* g) {
    asm volatile("global_load_async_to_lds_b128 %0, %1, off"
                 :: "v"(lds_byte_off), "v"(g)
                 : "memory");
}
__device__ __forceinline__ void wait_async0() {
    asm volatile("s_wait_asynccnt 0x0" ::: "memory");
}

// ---------------------------------------------------------------------------
// Kernel 1: per-edge MLP mask + A = adj*mask + I
// softmax([l0,l1])[1] == sigmoid(l1 - l0), so fold Wm2/bm2 into a difference.
// ---------------------------------------------------------------------------
__global__ void k_edge(const float* __restrict__ adj,
                       const float* __restrict__ xdeg,
                       const float* __restrict__ ydeg,
                       const float* __restrict__ Wm1,   // [3,16] row-major
                       const float* __restrict__ bm1,   // [16]
                       const float* __restrict__ Wm2,   // [16,2] row-major
                       const float* __restrict__ bm2,   // [2]
                       float* __restrict__ A) {
    __shared__ float sW1[48];   // Wm1
    __shared__ float sb1[16];   // bm1
    __shared__ float swd[16];   // Wm2[:,1]-Wm2[:,0]
    __shared__ float sbd;       // bm2[1]-bm2[0]
    const int tid = threadIdx.x;
    if (tid < 48) sW1[tid] = Wm1[tid];
    if (tid < 16) {
        sb1[tid] = bm1[tid];
        swd[tid] = Wm2[tid * 2 + 1] - Wm2[tid * 2 + 0];
    }
    if (tid == 0) sbd = bm2[1] - bm2[0];
    __syncthreads();

    const int idx = blockIdx.x * 256 + tid;      // < NN*NN (fits int)
    const int i = idx >> 11;                     // NN == 2048
    const int j = idx & (NN - 1);

    const float a  = adj[idx];
    const float xd = xdeg[idx];
    const float yd = ydeg[idx];

    float l = sbd;
#pragma unroll
    for (int k = 0; k < 16; ++k) {
        float h = a * sW1[k] + xd * sW1[16 + k] + yd * sW1[32 + k] + sb1[k];
        h = fmaxf(h, 0.0f);
        l += h * swd[k];
    }
    const float mask = 1.0f / (1.0f + __expf(-l));   // sigmoid(l1-l0)
    A[idx] = a * mask + ((i == j) ? 1.0f : 0.0f);
}

// ---------------------------------------------------------------------------
// Kernel 2: row sums -> dr = rowsum>0 ? rsqrt(rowsum) : 0   (block per row)
// ---------------------------------------------------------------------------
__global__ void k_rowsum(const float* __restrict__ A, float* __restrict__ dr) {
    const int i = blockIdx.x;
    const int tid = threadIdx.x;
    const float* Ai = A + (size_t)i * NN;
    float s = 0.0f;
    for (int j = tid; j < NN; j += 256) s += Ai[j];
    __shared__ float red[256];
    red[tid] = s;
    __syncthreads();
    for (int st = 128; st > 0; st >>= 1) {
        if (tid < st) red[tid] += red[tid + st];
        __syncthreads();
    }
    if (tid == 0) {
        const float t = red[0];
        dr[i] = (t > 0.0f) ? rsqrtf(t) : 0.0f;
    }
}

// ---------------------------------------------------------------------------
// Kernel 3: column sums -> dc   (thread per column; coalesced row scans)
// ---------------------------------------------------------------------------
__global__ void k_colsum(const float* __restrict__ A, float* __restrict__ dc) {
    const int j = blockIdx.x * 256 + threadIdx.x;
    float s = 0.0f;
    for (int i = 0; i < NN; ++i) s += A[(size_t)i * NN + j];
    dc[j] = (s > 0.0f) ? rsqrtf(s) : 0.0f;
}

// ---------------------------------------------------------------------------
// Kernel 4: A[i][j] *= dr[i]*dc[j]
// ---------------------------------------------------------------------------
__global__ void k_scale(float* __restrict__ A,
                        const float* __restrict__ dr,
                        const float* __restrict__ dc) {
    const int idx = blockIdx.x * 256 + threadIdx.x;
    const int i = idx >> 11;
    const int j = idx & (NN - 1);
    A[idx] *= dr[i] * dc[j];
}

// ---------------------------------------------------------------------------
// WMMA f32 GEMM:  C[M,N] = A[M,K] @ B[K,N] (+ bias[N]), all row-major f32.
// Block: 256 threads = 8 waves in a 4(M) x 2(N) grid -> 64x32 block tile.
// K staged through LDS in chunks of 32 with DOUBLE-BUFFERED async DMA
// (global_load_async_to_lds_b128) -> 8 x v_wmma_f32_16x16x4_f32 per chunk,
// one barrier per chunk, prefetch overlapped with the matrix pipe.
// Requires M%64==0, N%32==0, K%32==0 (true for all call sites).
// Fragment layouts per CDNA5 ISA 7.12.2:
//   A 16x4 : lane<16 -> {K0,K1} of row lane ; lane>=16 -> {K2,K3} of row lane-16
//   B 4x16 : lane<16 -> col lane of {K0,K1} ; lane>=16 -> col lane-16 of {K2,K3}
//   C/D    : vgpr v, lane<16 -> (M=v, N=lane) ; lane>=16 -> (M=8+v, N=lane-16)
// ---------------------------------------------------------------------------
__global__ void k_gemm_wmma(const float* __restrict__ A,
                            const float* __restrict__ B,
                            const float* __restrict__ bias,   // may be null
                            float* __restrict__ C,
                            int M, int N, int K) {
    // Row pitch 36 floats (144 B): 16B-aligned rows, LDS-bank-conflict free
    __shared__ float sA[2][64][36];
    __shared__ float sB[2][32][36];

    const int tid  = threadIdx.x;
    const int lane = tid & 31;
    const int wave = tid >> 5;
    const int wm   = wave >> 1;        // 0..3
    const int wn   = wave & 1;         // 0..1
    const int half = lane >> 4;        // 0: K0/K1 side, 1: K2/K3 side
    const int r    = lane & 15;

    const int m0 = blockIdx.y * 64;
    const int n0 = blockIdx.x * 32;

    // --- staging geometry: 3 async B128 per thread per K-chunk -------------
    const int srow  = tid >> 3;        // 0..31
    const int scol4 = tid & 7;         // 0..7 (float4 column within 32-wide tile)
    // global base pointers at k0 = 0 (advance by constants inside the loop)
    const float* gA0 = A + (size_t)(m0 + srow)      * K + scol4 * 4;
    const float* gA1 = A + (size_t)(m0 + srow + 32) * K + scol4 * 4;
    const float* gB  = B + (size_t)srow             * N + n0 + scol4 * 4;
    // LDS destination byte offsets (per buffer)
    uint32_t dA0[2], dA1[2], dB[2];
#pragma unroll
    for (int b = 0; b < 2; ++b) {
        dA0[b] = lds_off(&sA[b][srow][scol4 * 4]);
        dA1[b] = lds_off(&sA[b][srow + 32][scol4 * 4]);
        dB[b]  = lds_off(&sB[b][srow][scol4 * 4]);
    }

    v8f acc = {0.f, 0.f, 0.f, 0.f, 0.f, 0.f, 0.f, 0.f};

    // prefetch first tile into buffer 0
    async_ld_b128(dA0[0], gA0);
    async_ld_b128(dA1[0], gA1);
    async_ld_b128(dB[0],  gB);

    int buf = 0;
    for (int k0 = 0; k0 < K; k0 += 32) {
        wait_async0();        // this wave's async writes have landed in LDS
        __syncthreads();      // everyone's landed; prior readers of buf^1 done

        if (k0 + 32 < K) {    // prefetch next tile into the other buffer
            const int nk = k0 + 32;
            async_ld_b128(dA0[buf ^ 1], gA0 + nk);
            async_ld_b128(dA1[buf ^ 1], gA1 + nk);
            async_ld_b128(dB[buf ^ 1],  gB + (size_t)nk * N);
        }

#pragma unroll
        for (int kk = 0; kk < 32; kk += 4) {
            v2f a, b;
            a.x = sA[buf][wm * 16 + r][kk + half * 2 + 0];
            a.y = sA[buf][wm * 16 + r][kk + half * 2 + 1];
            b.x = sB[buf][kk + half * 2 + 0][wn * 16 + r];
            b.y = sB[buf][kk + half * 2 + 1][wn * 16 + r];
            acc = __builtin_amdgcn_wmma_f32_16x16x4_f32(
                /*neg_a=*/false, a, /*neg_b=*/false, b,
                /*c_mod=*/(short)0, acc, /*reuse_a=*/false, /*reuse_b=*/false);
        }
        buf ^= 1;
    }

    const int col = n0 + wn * 16 + r;
    const float bv = bias ? bias[col] : 0.0f;
#pragma unroll
    for (int v = 0; v < 8; ++v) {
        const int row = m0 + wm * 16 + half * 8 + v;
        C[(size_t)row * N + col] = acc[v] + bv;
    }
}

// ---------------------------------------------------------------------------
// Kernel: HG[i,0..3] = hid[i,:] @ Wg2   (thread per row; Wg2 in LDS)
// ---------------------------------------------------------------------------
__global__ void k_hg(const float* __restrict__ hid,
                     const float* __restrict__ Wg2,   // [64,4]
                     float* __restrict__ HG) {
    __shared__ float sw[H2 * NCLS];   // 256 floats
    const int tid = threadIdx.x;
    sw[tid] = Wg2[tid];
    __syncthreads();
    const int i = blockIdx.x * 256 + tid;
    const float* h = hid + (size_t)i * H2;
    float a0 = 0.f, a1 = 0.f, a2 = 0.f, a3 = 0.f;
#pragma unroll 8
    for (int k = 0; k < H2; ++k) {
        const float hv = h[k];
        a0 += hv * sw[k * 4 + 0];
        a1 += hv * sw[k * 4 + 1];
        a2 += hv * sw[k * 4 + 2];
        a3 += hv * sw[k * 4 + 3];
    }
    float* o = HG + (size_t)i * NCLS;
    o[0] = a0; o[1] = a1; o[2] = a2; o[3] = a3;
}

// ---------------------------------------------------------------------------
// Kernel: out[i,0..3] = A[i,:] @ HG   (block per row, LDS tree reduce)
// ---------------------------------------------------------------------------
__global__ void k_out(const float* __restrict__ A,
                      const float* __restrict__ HG,
                      float* __restrict__ out) {
    const int i = blockIdx.x;
    const int tid = threadIdx.x;
    const float* Ai = A + (size_t)i * NN;
    const v4f* hg4 = (const v4f*)HG;            // 16B-aligned ws carve
    v4f acc = {0.f, 0.f, 0.f, 0.f};
    for (int j = tid; j < NN; j += 256) {
        acc += Ai[j] * hg4[j];
    }
    __shared__ float red[256][5];   // pad to dodge bank conflicts
    red[tid][0] = acc.x; red[tid][1] = acc.y; red[tid][2] = acc.z; red[tid][3] = acc.w;
    __syncthreads();
    for (int st = 128; st > 0; st >>= 1) {
        if (tid < st) {
            red[tid][0] += red[tid + st][0];
            red[tid][1] += red[tid + st][1];
            red[tid][2] += red[tid + st][2];
            red[tid][3] += red[tid + st][3];
        }
        __syncthreads();
    }
    if (tid == 0) {
        float* o = out + (size_t)i * NCLS;
        o[0] = red[0][0]; o[1] = red[0][1]; o[2] = red[0][2]; o[3] = red[0][3];
    }
}

// ---------------------------------------------------------------------------
// Host-side orchestration
// ---------------------------------------------------------------------------
extern "C" void kernel_launch(void* const* d_in, const int* in_sizes, int n_in,
                              void* d_out, int out_size, void* d_ws, size_t ws_size,
                              hipStream_t stream) {
    (void)in_sizes; (void)n_in; (void)out_size; (void)ws_size;

    const float* x    = (const float*)d_in[0];
    const float* adj  = (const float*)d_in[1];
    const float* xdeg = (const float*)d_in[2];
    const float* ydeg = (const float*)d_in[3];
    const float* Wm1  = (const float*)d_in[4];
    const float* bm1  = (const float*)d_in[5];
    const float* Wm2  = (const float*)d_in[6];
    const float* bm2  = (const float*)d_in[7];
    const float* Wg1  = (const float*)d_in[8];
    const float* Wl2  = (const float*)d_in[9];
    const float* bl2  = (const float*)d_in[10];
    const float* Wg2  = (const float*)d_in[11];

    float* out = (float*)d_out;                       // [2048, 4]
    float* hid = (float*)d_out + (size_t)NN * NCLS;   // [2048, 64]

    // Workspace carve (floats): A | XW | H1 | HG | dr | dc   (~21 MB total)
    float* W  = (float*)d_ws;
    float* A  = W;
    float* XW = A  + (size_t)NN * NN;
    float* H1 = XW + (size_t)NN * HID;
    float* HG = H1 + (size_t)NN * HID;
    float* dr = HG + (size_t)NN * NCLS;
    float* dc = dr + NN;

    const dim3 blk(256);

    // 1) edge MLP mask -> A (unnormalized)
    k_edge<<<dim3(NN * NN / 256), blk, 0, stream>>>(adj, xdeg, ydeg,
                                                    Wm1, bm1, Wm2, bm2, A);
    // 2-4) degree normalization
    k_rowsum<<<dim3(NN), blk, 0, stream>>>(A, dr);
    k_colsum<<<dim3(NN / 256), blk, 0, stream>>>(A, dc);
    k_scale<<<dim3(NN * NN / 256), blk, 0, stream>>>(A, dr, dc);

    // 5) XW = x @ Wg1                [2048,1024] @ [1024,256]
    k_gemm_wmma<<<dim3(HID / 32, NN / 64), blk, 0, stream>>>(
        x, Wg1, nullptr, XW, NN, HID, IND);
    // 6) H1 = A @ XW                 [2048,2048] @ [2048,256]
    k_gemm_wmma<<<dim3(HID / 32, NN / 64), blk, 0, stream>>>(
        A, XW, nullptr, H1, NN, HID, NN);
    // 7) hid = H1 @ Wl2 + bl2        [2048,256] @ [256,64]  -> d_out tail
    k_gemm_wmma<<<dim3(H2 / 32, NN / 64), blk, 0, stream>>>(
        H1, Wl2, bl2, hid, NN, H2, HID);

    // 8) HG = hid @ Wg2              [2048,64] @ [64,4]
    k_hg<<<dim3(NN / 256), blk, 0, stream>>>(hid, Wg2, HG);
    // 9) out = A @ HG                [2048,2048] @ [2048,4] -> d_out head
    k_out<<<dim3(NN), blk, 0, stream>>>(A, HG, out);
}